// MentionDetector_36060545417613
// MI455X (gfx1250) — compile-verified
//
#include <hip/hip_runtime.h>

#define NMENT 8192
#define DIM   1536
#define TK    50
#define NEG_INF (-__builtin_inff())

#define KC 32          // K-chunk staged in LDS
#define BN 64          // block N tile (4 x 16 per wave)
#define BM 128         // block M tile (8 waves x 16 rows)
#define BSTRIDE 80     // padded LDS row stride (floats): conflict-free reads

typedef __attribute__((ext_vector_type(2))) float v2f;
typedef __attribute__((ext_vector_type(4))) float v4f;
typedef __attribute__((ext_vector_type(8))) float v8f;
typedef int v4i_gcc __attribute__((vector_size(16)));   // matches builtin param

#if __has_builtin(__builtin_amdgcn_global_load_async_to_lds_b128)
#define HAVE_ASYNC_LDS 1
#else
#define HAVE_ASYNC_LDS 0
#endif

__device__ __forceinline__ void async_copy16(const float* src, float* ldst) {
#if HAVE_ASYNC_LDS
  __builtin_amdgcn_global_load_async_to_lds_b128(
      (__attribute__((address_space(1))) v4i_gcc*)(src),
      (__attribute__((address_space(3))) v4i_gcc*)(ldst),
      0, 0);
#else
  *(v4f*)ldst = *(const v4f*)src;   // global_load_b128 + ds_store_b128
#endif
}

__device__ __forceinline__ void wait_async_keep2() {
#if HAVE_ASYNC_LDS
#if __has_builtin(__builtin_amdgcn_s_wait_asynccnt)
  __builtin_amdgcn_s_wait_asynccnt(2);
#else
  asm volatile("s_wait_asynccnt 2" ::: "memory");
#endif
#endif
}

__device__ __forceinline__ void wait_async_all() {
#if HAVE_ASYNC_LDS
#if __has_builtin(__builtin_amdgcn_s_wait_asynccnt)
  __builtin_amdgcn_s_wait_asynccnt(0);
#else
  asm volatile("s_wait_asynccnt 0" ::: "memory");
#endif
#endif
}

// C[M,Hd] = leaky_relu(A[M,K] @ W[K,Hd] + bias), fp32 WMMA 16x16x4.
// Block: 8 waves, 128x64 C tile; wave w -> rows [w*16, w*16+16), all 64 cols.
// B panel (KCx64) staged in LDS via async global->LDS, double buffered.
__global__ __launch_bounds__(256) void gemm_lrelu_kernel(
    const float* __restrict__ A, const float* __restrict__ W,
    const float* __restrict__ bias, float* __restrict__ C,
    int M, int K, int Hd) {
  __shared__ float Bs[2][KC][BSTRIDE];

  const int tid   = threadIdx.x;
  const int wave  = tid >> 5;
  const int lane  = tid & 31;
  const int n16   = lane & 15;
  const int khalf = (lane >> 4) << 1;   // 0 or 2

  const int row0 = blockIdx.y * BM + wave * 16;
  const int col0 = blockIdx.x * BN;

  // A-frag source: lane holds A[row0 + (lane&15)][k + khalf .. +1]
  const float* Arow = A + (size_t)(row0 + n16) * K + khalf;

  // B staging coords: thread covers panel rows (srow, srow+16), 4 floats at scol
  const int srow = tid >> 4;            // 0..15
  const int scol = (tid & 15) << 2;     // 0,4,..,60
  const float* Wst = W + (size_t)srow * Hd + col0 + scol;

  v8f acc[4];
#pragma unroll
  for (int t = 0; t < 4; ++t) acc[t] = (v8f){0.f,0.f,0.f,0.f,0.f,0.f,0.f,0.f};

  // Prologue: stage panel 0 into buffer 0 (2 async b128 per thread)
  async_copy16(Wst, &Bs[0][srow][scol]);
  async_copy16(Wst + (size_t)16 * Hd, &Bs[0][srow + 16][scol]);

  int buf = 0;
  for (int kc = 0; kc < K; kc += KC, buf ^= 1) {
    if (kc + KC < K) {  // stage next panel into the other buffer, then wait
      const float* Wn = Wst + (size_t)(kc + KC) * Hd;
      async_copy16(Wn, &Bs[buf ^ 1][srow][scol]);
      async_copy16(Wn + (size_t)16 * Hd, &Bs[buf ^ 1][srow + 16][scol]);
      wait_async_keep2();   // in-order: current panel complete, next in flight
    } else {
      wait_async_all();
    }
    __syncthreads();

    __builtin_prefetch(Arow + kc + KC, 0, 1);  // global_prefetch_b8 (A stream)

#pragma unroll
    for (int k = 0; k < KC; k += 4) {
      v2f a = *(const v2f*)(Arow + kc + k);              // aligned b64
      const float* brow0 = &Bs[buf][k + khalf][n16];
      const float* brow1 = &Bs[buf][k + khalf + 1][n16];
#pragma unroll
      for (int t = 0; t < 4; ++t) {
        v2f b = { brow0[16 * t], brow1[16 * t] };
        acc[t] = __builtin_amdgcn_wmma_f32_16x16x4_f32(
            /*neg_a=*/false, a, /*neg_b=*/false, b,
            /*c_mod=*/(short)0, acc[t], /*reuse_a=*/false, /*reuse_b=*/false);
      }
    }
    __syncthreads();  // everyone done with buf before it is restaged
  }

  // C/D layout: VGPR r -> M = r + 8*(lane>=16), N = lane&15
  const int mbase = row0 + ((lane >> 4) << 3);
#pragma unroll
  for (int t = 0; t < 4; ++t) {
    const int n = col0 + 16 * t + n16;
    const float bn = bias[n];
#pragma unroll
    for (int r = 0; r < 8; ++r) {
      float v = acc[t][r] + bn;
      v = (v > 0.f) ? v : 0.01f * v;
      C[(size_t)(mbase + r) * Hd + n] = v;
    }
  }
}

// s[i] = dot(H2[i,:], Wout) + bout  (one 256-thread block per row)
__global__ __launch_bounds__(256) void score_kernel(
    const float* __restrict__ H2, const float* __restrict__ Wout,
    const float* __restrict__ bout, float* __restrict__ s, int Hd) {
  __shared__ float red[256];
  const int i = blockIdx.x;
  const float* hrow = H2 + (size_t)i * Hd;
  float acc = 0.f;
  for (int j = threadIdx.x; j < Hd; j += 256) acc += hrow[j] * Wout[j];
  red[threadIdx.x] = acc;
  __syncthreads();
  for (int off = 128; off > 0; off >>= 1) {
    if (threadIdx.x < off) red[threadIdx.x] += red[threadIdx.x + off];
    __syncthreads();
  }
  if (threadIdx.x == 0) s[i] = red[0] + bout[0];
}

// Row i: top-50 of { s[i]+s[j] : j<i } desc, lower-index tie-break; pad slots
// t>=i with (-inf, t) to match jax.lax.top_k over the -inf masked region.
__global__ __launch_bounds__(256) void topk_kernel(
    const float* __restrict__ s, float* __restrict__ out_scores,
    float* __restrict__ out_idx, int n) {
  __shared__ float vals[NMENT];
  __shared__ float rv[256];
  __shared__ int   ri[256];
  const int i   = blockIdx.x;
  const int tid = threadIdx.x;
  const float si = s[i];
  for (int j = tid; j < i; j += 256) vals[j] = si + s[j];
  __syncthreads();

  for (int t = 0; t < TK; ++t) {
    float best = NEG_INF;
    int bidx = 0x7fffffff;
    for (int j = tid; j < i; j += 256) {
      float v = vals[j];
      if (v > best) { best = v; bidx = j; }   // strict '>' keeps lowest j
    }
    rv[tid] = best; ri[tid] = bidx;
    __syncthreads();
    for (int off = 128; off > 0; off >>= 1) {
      if (tid < off) {
        float v2 = rv[tid + off]; int i2 = ri[tid + off];
        if (v2 > rv[tid] || (v2 == rv[tid] && i2 < ri[tid])) {
          rv[tid] = v2; ri[tid] = i2;
        }
      }
      __syncthreads();
    }
    if (tid == 0) {
      if (t < i) {
        out_scores[(size_t)i * TK + t] = rv[0];
        out_idx  [(size_t)i * TK + t] = (float)ri[0];
        vals[ri[0]] = NEG_INF;               // extract winner
      } else {
        out_scores[(size_t)i * TK + t] = NEG_INF;
        out_idx  [(size_t)i * TK + t] = (float)t;  // pad index == slot
      }
    }
    __syncthreads();
  }
}

extern "C" void kernel_launch(void* const* d_in, const int* in_sizes, int n_in,
                              void* d_out, int out_size, void* d_ws, size_t ws_size,
                              hipStream_t stream) {
  const float* mentions = (const float*)d_in[0];
  const float* W0   = (const float*)d_in[1];
  const float* b0   = (const float*)d_in[2];
  const float* W1   = (const float*)d_in[3];
  const float* b1   = (const float*)d_in[4];
  const float* Wout = (const float*)d_in[5];
  const float* bout = (const float*)d_in[6];
  // d_in[7] is k (=50); fixed at compile time as TK.

  float* H1 = (float*)d_ws;                       // [NMENT, DIM]
  float* H2 = H1 + (size_t)NMENT * DIM;           // [NMENT, DIM]
  float* sS = H2 + (size_t)NMENT * DIM;           // [NMENT]

  float* out_scores = (float*)d_out;                    // [NMENT, TK]
  float* out_idx    = out_scores + (size_t)NMENT * TK;  // [NMENT, TK] (as float)

  dim3 grid(DIM / BN, NMENT / BM);  // 24 x 64
  gemm_lrelu_kernel<<<grid, 256, 0, stream>>>(mentions, W0, b0, H1, NMENT, DIM, DIM);
  gemm_lrelu_kernel<<<grid, 256, 0, stream>>>(H1, W1, b1, H2, NMENT, DIM, DIM);
  score_kernel<<<NMENT, 256, 0, stream>>>(H2, Wout, bout, sS, DIM);
  topk_kernel<<<NMENT, 256, 0, stream>>>(sS, out_scores, out_idx, NMENT);
}